// MotionCorrect_54494545051840
// MI455X (gfx1250) — compile-verified
//
#include <hip/hip_runtime.h>
#include <math.h>

typedef float v2f __attribute__((ext_vector_type(2)));
typedef float v8f __attribute__((ext_vector_type(8)));

#define NXD 512
#define NYD 512
#define NZD 32
#define NXY (NXD * NYD)            // 262144
#define NTOT ((size_t)NXY * NZD)   // 8388608
#define TWO_PI 6.283185307179586f

static __device__ inline v2f v2neg(v2f a) { v2f r; r.x = -a.x; r.y = -a.y; return r; }

#if __has_builtin(__builtin_amdgcn_wmma_f32_16x16x4_f32)
// f32 WMMA has no A/B NEG modifier (ISA: NEG = {CNeg,0,0} for F32) -> negate A in VALU
#define WMMA_ACC(A, B, C)  __builtin_amdgcn_wmma_f32_16x16x4_f32(false, (A), false, (B), (short)0, (C), false, false)
#define WMMA_NEGA(A, B, C) __builtin_amdgcn_wmma_f32_16x16x4_f32(false, v2neg(A), false, (B), (short)0, (C), false, false)
#else
// fallback so the round still compiles; histogram (wmma==0) will reveal this path was taken
static __device__ inline v8f wmma_fallback(v2f a, v2f b, v8f c) { c[0] += a.x * b.x; return c; }
static __device__ inline v8f wmma_fallback_n(v2f a, v2f b, v8f c) { c[0] -= a.x * b.x; return c; }
#define WMMA_ACC(A, B, C)  wmma_fallback((A), (B), (C))
#define WMMA_NEGA(A, B, C) wmma_fallback_n((A), (B), (C))
#endif

#define V8Z {0.f,0.f,0.f,0.f,0.f,0.f,0.f,0.f}

// ---------------------------------------------------------------------------
// small reductions
// ---------------------------------------------------------------------------
__global__ void init_scalars_k(float* s) {
    if (threadIdx.x < 4) s[threadIdx.x] = 0.0f;
}

__global__ void reduce_sum_k(const float* __restrict__ in, size_t n, float* __restrict__ out) {
    __shared__ float sm[256];
    float acc = 0.f;
    for (size_t i = (size_t)blockIdx.x * 256 + threadIdx.x; i < n; i += (size_t)gridDim.x * 256)
        acc += in[i];
    sm[threadIdx.x] = acc;
    __syncthreads();
    for (int w = 128; w > 0; w >>= 1) {
        if ((int)threadIdx.x < w) sm[threadIdx.x] += sm[threadIdx.x + w];
        __syncthreads();
    }
    if (threadIdx.x == 0) atomicAdd(out, sm[0]);
}

__global__ void reduce_sumsq_zm_k(const float* __restrict__ in, size_t n,
                                  const float* __restrict__ sumPtr, float invN,
                                  float* __restrict__ out) {
    __shared__ float sm[256];
    float mean = sumPtr[0] * invN;
    float acc = 0.f;
    for (size_t i = (size_t)blockIdx.x * 256 + threadIdx.x; i < n; i += (size_t)gridDim.x * 256) {
        float d = in[i] - mean;
        acc += d * d;
    }
    sm[threadIdx.x] = acc;
    __syncthreads();
    for (int w = 128; w > 0; w >>= 1) {
        if ((int)threadIdx.x < w) sm[threadIdx.x] += sm[threadIdx.x + w];
        __syncthreads();
    }
    if (threadIdx.x == 0) atomicAdd(out, sm[0]);
}

// ---------------------------------------------------------------------------
// forward DFT-32 along Z (real -> complex), zero-mean applied, via WMMA matmul
// one wave handles 16 contiguous (x,y) columns: out(32x16) = W32(32x32) * X(32x16)
// ---------------------------------------------------------------------------
__global__ __launch_bounds__(256) void zdft_fwd_k(const float* __restrict__ in,
                                                  float2* __restrict__ out,
                                                  const float* __restrict__ sumPtr,
                                                  float invN) {
    __shared__ float lsh[8][512];
    const int lane = threadIdx.x & 31;
    const int wv   = threadIdx.x >> 5;
    const int col0 = (blockIdx.x * 8 + wv) * 16;
    const float mean = sumPtr[0] * invN;

    float* l = lsh[wv];
    const float* src = in + (size_t)col0 * NZD;
#pragma unroll
    for (int i = 0; i < 16; ++i) l[lane + 32 * i] = src[lane + 32 * i] - mean;
    __syncthreads();

    const int c  = lane & 15;
    const int hi = lane >> 4;
    const int m0 = lane & 15;

    v8f re0 = V8Z, re1 = V8Z, im0 = V8Z, im1 = V8Z;
#pragma unroll
    for (int k0 = 0; k0 < 32; k0 += 4) {
        const int ka = k0 + 2 * hi;
        v2f b; b.x = l[c * 32 + ka]; b.y = l[c * 32 + ka + 1];
        const float th0 = (TWO_PI / 32.f) * (float)((m0 * ka) & 31);
        const float th1 = (TWO_PI / 32.f) * (float)((m0 * (ka + 1)) & 31);
        const float th2 = (TWO_PI / 32.f) * (float)(((m0 + 16) * ka) & 31);
        const float th3 = (TWO_PI / 32.f) * (float)(((m0 + 16) * (ka + 1)) & 31);
        v2f aR0, aI0, aR1, aI1;
        aR0.x = __cosf(th0);  aR0.y = __cosf(th1);
        aI0.x = -__sinf(th0); aI0.y = -__sinf(th1);     // forward: e^{-i th}
        aR1.x = __cosf(th2);  aR1.y = __cosf(th3);
        aI1.x = -__sinf(th2); aI1.y = -__sinf(th3);
        re0 = WMMA_ACC(aR0, b, re0);
        re1 = WMMA_ACC(aR1, b, re1);
        im0 = WMMA_ACC(aI0, b, im0);
        im1 = WMMA_ACC(aI1, b, im1);
    }

    float2* dst = out + (size_t)(col0 + c) * NZD;
#pragma unroll
    for (int v = 0; v < 8; ++v) {
        const int z0 = v + 8 * hi;           // C layout: row = vgpr (+8 for hi lanes)
        dst[z0]      = make_float2(re0[v], im0[v]);
        dst[z0 + 16] = make_float2(re1[v], im1[v]);
    }
}

// ---------------------------------------------------------------------------
// 512-point complex FFT (in-place along a strided line), four-step 512 = 32*16
// one wave per line; both 32-DFT and 16-DFT stages are WMMA matmuls.
// sgn = -1 forward, +1 inverse (unscaled).
// ---------------------------------------------------------------------------
__global__ __launch_bounds__(256) void fft512_k(float2* __restrict__ buf,
                                                int outerStride, int estride, float sgn) {
    __shared__ float2 lsh[8][512];
    const int lane = threadIdx.x & 31;
    const int wv   = threadIdx.x >> 5;
    const int line = blockIdx.x * 8 + wv;        // 16384 lines
    const int oi = line >> 5;
    const int zi = line & 31;
    const size_t base = (size_t)oi * outerStride + zi;
    float2* l = lsh[wv];

    for (int i = 0; i < 16; ++i) {
        const int e = lane + 32 * i;
        l[e] = buf[base + (size_t)e * estride];
    }
    __syncthreads();

    const int c  = lane & 15;
    const int hi = lane >> 4;
    const int m0 = lane & 15;

    // ---- step 1: T = W32 x M   (M[n1][n2] = l[n1*16 + n2], complex) ----
    v8f re0 = V8Z, re1 = V8Z, im0 = V8Z, im1 = V8Z;
#pragma unroll
    for (int k0 = 0; k0 < 32; k0 += 4) {
        const int ka = k0 + 2 * hi;
        const float2 e0 = l[ka * 16 + c];
        const float2 e1 = l[(ka + 1) * 16 + c];
        v2f br, bi;
        br.x = e0.x; br.y = e1.x;
        bi.x = e0.y; bi.y = e1.y;
        const float th0 = (TWO_PI / 32.f) * (float)((m0 * ka) & 31);
        const float th1 = (TWO_PI / 32.f) * (float)((m0 * (ka + 1)) & 31);
        const float th2 = (TWO_PI / 32.f) * (float)(((m0 + 16) * ka) & 31);
        const float th3 = (TWO_PI / 32.f) * (float)(((m0 + 16) * (ka + 1)) & 31);
        v2f aR0, aI0, aR1, aI1;
        aR0.x = __cosf(th0);       aR0.y = __cosf(th1);
        aI0.x = sgn * __sinf(th0); aI0.y = sgn * __sinf(th1);
        aR1.x = __cosf(th2);       aR1.y = __cosf(th3);
        aI1.x = sgn * __sinf(th2); aI1.y = sgn * __sinf(th3);
        // Re += Wr*Mr - Wi*Mi ; Im += Wr*Mi + Wi*Mr
        re0 = WMMA_ACC(aR0, br, re0); re0 = WMMA_NEGA(aI0, bi, re0);
        im0 = WMMA_ACC(aR0, bi, im0); im0 = WMMA_ACC(aI0, br, im0);
        re1 = WMMA_ACC(aR1, br, re1); re1 = WMMA_NEGA(aI1, bi, re1);
        im1 = WMMA_ACC(aR1, bi, im1); im1 = WMMA_ACC(aI1, br, im1);
    }
    __syncthreads();

    // ---- step 2: twiddle T'[k1][n2] = T * e^{sgn*i*2pi*k1*n2/512}, back to LDS ----
    const int n2 = lane & 15;
#pragma unroll
    for (int v = 0; v < 8; ++v) {
        const int k1 = v + 8 * hi;
        {
            const float th = (TWO_PI / 512.f) * (float)((k1 * n2) & 511);
            const float cr = __cosf(th), ci = sgn * __sinf(th);
            const float xr = re0[v], xi = im0[v];
            l[k1 * 16 + n2] = make_float2(xr * cr - xi * ci, xr * ci + xi * cr);
        }
        {
            const int k1b = k1 + 16;
            const float th = (TWO_PI / 512.f) * (float)((k1b * n2) & 511);
            const float cr = __cosf(th), ci = sgn * __sinf(th);
            const float yr = re1[v], yi = im1[v];
            l[k1b * 16 + n2] = make_float2(yr * cr - yi * ci, yr * ci + yi * cr);
        }
    }
    __syncthreads();

    // ---- step 3: Out[k1][k2] = sum_n2 T'[k1][n2] * e^{sgn*i*2pi*n2*k2/16} ----
    v8f Re0 = V8Z, Re1 = V8Z, Im0 = V8Z, Im1 = V8Z;
#pragma unroll
    for (int k0 = 0; k0 < 16; k0 += 4) {
        const int ka = k0 + 2 * hi;
        const float2 a00 = l[m0 * 16 + ka],        a01 = l[m0 * 16 + ka + 1];
        const float2 a10 = l[(m0 + 16) * 16 + ka], a11 = l[(m0 + 16) * 16 + ka + 1];
        v2f Ar0, Ai0, Ar1, Ai1;
        Ar0.x = a00.x; Ar0.y = a01.x;  Ai0.x = a00.y; Ai0.y = a01.y;
        Ar1.x = a10.x; Ar1.y = a11.x;  Ai1.x = a10.y; Ai1.y = a11.y;
        const float tb0 = (TWO_PI / 16.f) * (float)((ka * c) & 15);
        const float tb1 = (TWO_PI / 16.f) * (float)(((ka + 1) * c) & 15);
        v2f Br, Bi;
        Br.x = __cosf(tb0);       Br.y = __cosf(tb1);
        Bi.x = sgn * __sinf(tb0); Bi.y = sgn * __sinf(tb1);
        Re0 = WMMA_ACC(Ar0, Br, Re0); Re0 = WMMA_NEGA(Ai0, Bi, Re0);
        Im0 = WMMA_ACC(Ar0, Bi, Im0); Im0 = WMMA_ACC(Ai0, Br, Im0);
        Re1 = WMMA_ACC(Ar1, Br, Re1); Re1 = WMMA_NEGA(Ai1, Bi, Re1);
        Im1 = WMMA_ACC(Ar1, Bi, Im1); Im1 = WMMA_ACC(Ai1, Br, Im1);
    }

    // out index k = k1 + 32*k2 ; k2 = c, k1 = v + 8*hi (+16)
#pragma unroll
    for (int v = 0; v < 8; ++v) {
        const int k1 = v + 8 * hi;
        buf[base + (size_t)(k1 + 32 * c) * estride]      = make_float2(Re0[v], Im0[v]);
        buf[base + (size_t)(k1 + 16 + 32 * c) * estride] = make_float2(Re1[v], Im1[v]);
    }
}

// ---------------------------------------------------------------------------
// pointwise imgF *= conj(tplF)
// ---------------------------------------------------------------------------
__global__ void crossmul_k(float2* __restrict__ img, const float2* __restrict__ tpl, size_t n) {
    for (size_t i = (size_t)blockIdx.x * blockDim.x + threadIdx.x; i < n;
         i += (size_t)gridDim.x * blockDim.x) {
        const float2 a = img[i], b = tpl[i];
        img[i] = make_float2(a.x * b.x + a.y * b.y, a.y * b.x - a.x * b.y);
    }
}

// ---------------------------------------------------------------------------
// inverse DFT-32 along Z (complex -> |.|), stores rolled by (256,256,16)
// ---------------------------------------------------------------------------
__global__ __launch_bounds__(256) void zidft_abs_rolled_k(const float2* __restrict__ in,
                                                          float* __restrict__ outRolled,
                                                          float scale) {
    __shared__ float2 lsh[8][512];
    const int lane = threadIdx.x & 31;
    const int wv   = threadIdx.x >> 5;
    const int col0 = (blockIdx.x * 8 + wv) * 16;
    float2* l = lsh[wv];

    const float2* src = in + (size_t)col0 * NZD;
#pragma unroll
    for (int i = 0; i < 16; ++i) l[lane + 32 * i] = src[lane + 32 * i];
    __syncthreads();

    const int c  = lane & 15;
    const int hi = lane >> 4;
    const int m0 = lane & 15;

    v8f re0 = V8Z, re1 = V8Z, im0 = V8Z, im1 = V8Z;
#pragma unroll
    for (int k0 = 0; k0 < 32; k0 += 4) {
        const int ka = k0 + 2 * hi;
        const float2 e0 = l[c * 32 + ka];
        const float2 e1 = l[c * 32 + ka + 1];
        v2f br, bi;
        br.x = e0.x; br.y = e1.x;
        bi.x = e0.y; bi.y = e1.y;
        const float th0 = (TWO_PI / 32.f) * (float)((m0 * ka) & 31);
        const float th1 = (TWO_PI / 32.f) * (float)((m0 * (ka + 1)) & 31);
        const float th2 = (TWO_PI / 32.f) * (float)(((m0 + 16) * ka) & 31);
        const float th3 = (TWO_PI / 32.f) * (float)(((m0 + 16) * (ka + 1)) & 31);
        v2f aR0, aI0, aR1, aI1;                       // inverse: e^{+i th}
        aR0.x = __cosf(th0); aR0.y = __cosf(th1);
        aI0.x = __sinf(th0); aI0.y = __sinf(th1);
        aR1.x = __cosf(th2); aR1.y = __cosf(th3);
        aI1.x = __sinf(th2); aI1.y = __sinf(th3);
        re0 = WMMA_ACC(aR0, br, re0); re0 = WMMA_NEGA(aI0, bi, re0);
        im0 = WMMA_ACC(aR0, bi, im0); im0 = WMMA_ACC(aI0, br, im0);
        re1 = WMMA_ACC(aR1, br, re1); re1 = WMMA_NEGA(aI1, bi, re1);
        im1 = WMMA_ACC(aR1, bi, im1); im1 = WMMA_ACC(aI1, br, im1);
    }

    const int col = col0 + c;
    const int x = col >> 9, y = col & 511;
    const int xo = (x + 256) & 511, yo = (y + 256) & 511;
    float* dst = outRolled + ((size_t)xo * NYD + yo) * NZD;
#pragma unroll
    for (int v = 0; v < 8; ++v) {
        const int z0 = v + 8 * hi;
        dst[(z0 + 16) & 31] = scale * sqrtf(re0[v] * re0[v] + im0[v] * im0[v]);
        dst[(z0 + 32) & 31] = scale * sqrtf(re1[v] * re1[v] + im1[v] * im1[v]);
    }
}

// ---------------------------------------------------------------------------
// summed-area tables of img and img^2 (3 directional passes)
// ---------------------------------------------------------------------------
__global__ void sat_z_k(const float* __restrict__ img, float* __restrict__ s1, float* __restrict__ s2) {
    const int col = blockIdx.x * blockDim.x + threadIdx.x;
    if (col >= NXY) return;
    const size_t base = (size_t)col * NZD;
    float a = 0.f, b = 0.f;
    for (int z = 0; z < NZD; ++z) {
        const float v = img[base + z];
        a += v; b += v * v;
        s1[base + z] = a; s2[base + z] = b;
    }
}
__global__ void sat_y_k(float* __restrict__ s1, float* __restrict__ s2) {
    const int id = blockIdx.x * blockDim.x + threadIdx.x;
    if (id >= NXD * NZD) return;
    const int x = id >> 5, z = id & 31;
    const size_t base = (size_t)x * NYD * NZD + z;
    float a = 0.f, b = 0.f;
    for (int y = 0; y < NYD; ++y) {
        const size_t p = base + (size_t)y * NZD;
        a += s1[p]; s1[p] = a;
        b += s2[p]; s2[p] = b;
    }
}
__global__ void sat_x_k(float* __restrict__ s1, float* __restrict__ s2) {
    const int id = blockIdx.x * blockDim.x + threadIdx.x;
    if (id >= NYD * NZD) return;
    const int y = id >> 5, z = id & 31;
    const size_t base = (size_t)y * NZD + z;
    float a = 0.f, b = 0.f;
    for (int x = 0; x < NXD; ++x) {
        const size_t p = base + (size_t)x * NYD * NZD;
        a += s1[p]; s1[p] = a;
        b += s2[p]; s2[p] = b;
    }
}

__device__ inline float sat_prefix(const float* __restrict__ S, int a, int b, int c) {
    if (a <= 0 || b <= 0 || c <= 0) return 0.f;
    return S[((size_t)(a - 1) * NYD + (b - 1)) * NZD + (c - 1)];
}
__device__ inline float win_sum(const float* __restrict__ S, int i, int j, int k,
                                int wx, int wy, int wz) {
    return sat_prefix(S, i + wx, j + wy, k + wz) - sat_prefix(S, i, j + wy, k + wz)
         - sat_prefix(S, i + wx, j, k + wz)      - sat_prefix(S, i + wx, j + wy, k)
         + sat_prefix(S, i, j, k + wz) + sat_prefix(S, i, j + wy, k)
         + sat_prefix(S, i + wx, j, k) - sat_prefix(S, i, j, k);
}

// ---------------------------------------------------------------------------
// ncc = rolled_cc window / sqrt(tpl_var * imgs_var)
// ---------------------------------------------------------------------------
__global__ void ncc_k(const float* __restrict__ s1, const float* __restrict__ s2,
                      const float* __restrict__ rolled, const float* __restrict__ tplVarPtr,
                      float* __restrict__ nccOut) {
    const int t = blockIdx.x * blockDim.x + threadIdx.x;
    if (t >= 21 * 21 * 5) return;
    const int i = t / 105, j = (t % 105) / 5, k = t % 5;
    const int wx = NXD - 20, wy = NYD - 20, wz = NZD - 4;
    const float invW = 1.f / ((float)wx * (float)wy * (float)wz);
    const float m1 = win_sum(s1, i, j, k, wx, wy, wz) * invW;
    const float m2 = win_sum(s2, i, j, k, wx, wy, wz) * invW;
    float var = m2 - (m1 * m1) / (float)NTOT + 1e-8f;
    if (var < 0.f) var = 0.f;
    const float den = sqrtf((tplVarPtr[0] + 1e-8f) * var);
    float v = rolled[((size_t)(246 + i) * NYD + (246 + j)) * NZD + (14 + k)] / den;
    if (v != v) v = 0.f;   // NaN -> 0
    nccOut[t] = v;
}

// ---------------------------------------------------------------------------
// argmax over (21,21,5) + Gaussian log sub-pixel refinement
// ---------------------------------------------------------------------------
__global__ void argmax_refine_k(const float* __restrict__ ncc, float* __restrict__ out) {
    __shared__ float vm[256];
    __shared__ int vi[256];
    const int t = threadIdx.x;
    float mv = -3.4e38f; int mi = 0;
    for (int i = t; i < 21 * 21 * 5; i += 256) {
        const float v = ncc[i];
        if (v > mv) { mv = v; mi = i; }
    }
    vm[t] = mv; vi[t] = mi;
    __syncthreads();
    for (int w = 128; w > 0; w >>= 1) {
        if (t < w && vm[t + w] > vm[t]) { vm[t] = vm[t + w]; vi[t] = vi[t + w]; }
        __syncthreads();
    }
    if (t == 0) {
        const int idx = vi[0];
        const int sx = idx / 105, sy = (idx % 105) / 5, sz = idx % 5;
        auto V = [&](int dx, int dy, int dz) {
            int a = sx + dx; a = a < 0 ? 0 : (a > 20 ? 20 : a);
            int b = sy + dy; b = b < 0 ? 0 : (b > 20 ? 20 : b);
            int c = sz + dz; c = c < 0 ? 0 : (c > 4 ? 4 : c);
            return __logf(ncc[a * 105 + b * 5 + c]);
        };
        const float six = 6.f * V(0, 0, 0);
        out[0] = -(float)(sx - 10) - (V(-1,0,0) - V(1,0,0)) / (2.f*V(-1,0,0) - six + 2.f*V(1,0,0));
        out[1] = -(float)(sy - 10) - (V(0,-1,0) - V(0,1,0)) / (2.f*V(0,-1,0) - six + 2.f*V(0,1,0));
        out[2] = -(float)(sz - 2)  - (V(0,0,-1) - V(0,0,1)) / (2.f*V(0,0,-1) - six + 2.f*V(0,0,1));
    }
}

// ---------------------------------------------------------------------------
extern "C" void kernel_launch(void* const* d_in, const int* in_sizes, int n_in,
                              void* d_out, int out_size, void* d_ws, size_t ws_size,
                              hipStream_t stream) {
    (void)in_sizes; (void)n_in; (void)out_size; (void)ws_size;
    const float* fr  = (const float*)d_in[0];
    const float* tpl = (const float*)d_in[1];
    float* out = (float*)d_out;
    char* ws = (char*)d_ws;

    float2* tplF = (float2*)ws;                            // 64 MB
    float2* imgF = (float2*)(ws + NTOT * sizeof(float2));  // 64 MB
    float*  S1   = (float*)(ws + 2 * NTOT * sizeof(float2));
    float*  S2   = S1 + NTOT;
    float*  scal = S2 + NTOT;   // [0]=tplSum [1]=imgSum [2]=tplVarSum

    float* rolled = out + 3;              // (512,512,32)
    float* nccOut = out + 3 + NTOT;       // (21,21,5)
    const float invN = 1.f / (float)NTOT;

    init_scalars_k<<<1, 32, 0, stream>>>(scal);
    reduce_sum_k<<<512, 256, 0, stream>>>(tpl, NTOT, scal + 0);
    reduce_sum_k<<<512, 256, 0, stream>>>(fr,  NTOT, scal + 1);
    reduce_sumsq_zm_k<<<512, 256, 0, stream>>>(tpl, NTOT, scal + 0, invN, scal + 2);

    // forward 3D FFT = DFT32(z) . FFT512(y) . FFT512(x)   (all WMMA matmul based)
    zdft_fwd_k<<<2048, 256, 0, stream>>>(tpl, tplF, scal + 0, invN);
    zdft_fwd_k<<<2048, 256, 0, stream>>>(fr,  imgF, scal + 1, invN);
    fft512_k<<<2048, 256, 0, stream>>>(tplF, NYD * NZD, NZD,       -1.f);  // y
    fft512_k<<<2048, 256, 0, stream>>>(imgF, NYD * NZD, NZD,       -1.f);
    fft512_k<<<2048, 256, 0, stream>>>(tplF, NZD,       NYD * NZD, -1.f);  // x
    fft512_k<<<2048, 256, 0, stream>>>(imgF, NZD,       NYD * NZD, -1.f);

    crossmul_k<<<2048, 256, 0, stream>>>(imgF, tplF, NTOT);

    // inverse (scale 1/N folded into the z-stage abs)
    fft512_k<<<2048, 256, 0, stream>>>(imgF, NZD,       NYD * NZD, +1.f);  // x
    fft512_k<<<2048, 256, 0, stream>>>(imgF, NYD * NZD, NZD,       +1.f);  // y
    zidft_abs_rolled_k<<<2048, 256, 0, stream>>>(imgF, rolled, invN);

    // window statistics for the NCC denominator
    sat_z_k<<<(NXY + 255) / 256, 256, 0, stream>>>(fr, S1, S2);
    sat_y_k<<<(NXD * NZD + 255) / 256, 256, 0, stream>>>(S1, S2);
    sat_x_k<<<(NYD * NZD + 255) / 256, 256, 0, stream>>>(S1, S2);

    ncc_k<<<(21 * 21 * 5 + 255) / 256, 256, 0, stream>>>(S1, S2, rolled, scal + 2, nccOut);
    argmax_refine_k<<<1, 256, 0, stream>>>(nccOut, out);
}